// GIN_5686536700272
// MI455X (gfx1250) — compile-verified
//
#include <hip/hip_runtime.h>
#include <hip/hip_bf16.h>

typedef __attribute__((ext_vector_type(2))) float v2f;
typedef __attribute__((ext_vector_type(8))) float v8f;

#define N_NODES_C 100000
#define N_EDGES_C 1600000
#define DIM_C     128
#define BN_EPS_C  1e-5f

// ---------------------------------------------------------------------------
// Zero-fill (float4 grid-stride)
// ---------------------------------------------------------------------------
__global__ void gin_zero_kernel(float* __restrict__ p, size_t n4) {
    size_t i = (size_t)blockIdx.x * blockDim.x + threadIdx.x;
    size_t stride = (size_t)gridDim.x * blockDim.x;
    float4* p4 = (float4*)p;
    float4 z = make_float4(0.f, 0.f, 0.f, 0.f);
    for (; i < n4; i += stride) p4[i] = z;
}

// ---------------------------------------------------------------------------
// Edge scatter-add: one wave (32 lanes) per edge, 128-dim features.
// Each lane: float4 gather of x[src], 4x global_atomic_add_f32 into agg[dst].
// ---------------------------------------------------------------------------
__global__ __launch_bounds__(256)
void gin_scatter_kernel(const float* __restrict__ x, const int* __restrict__ ei,
                        float* __restrict__ agg, int nEdges) {
    int gid  = blockIdx.x * blockDim.x + threadIdx.x;
    int edge = gid >> 5;
    int lane = gid & 31;
    if (edge >= nEdges) return;
    int s = ei[edge];
    int d = ei[nEdges + edge];
    const float4 v = *(const float4*)(x + (size_t)s * DIM_C + lane * 4);
    float* base = agg + (size_t)d * DIM_C + lane * 4;
    unsafeAtomicAdd(base + 0, v.x);   // global_atomic_add_f32
    unsafeAtomicAdd(base + 1, v.y);
    unsafeAtomicAdd(base + 2, v.z);
    unsafeAtomicAdd(base + 3, v.w);
}

// ---------------------------------------------------------------------------
// Fused GIN MLP: h=(x+agg); h@W1+b1 -> BN -> ReLU -> @W2+b2 -> ReLU
// 2 waves/block; each wave owns TWO 16-node tiles so every B-fragment load
// feeds two v_wmma_f32_16x16x4_f32 (A-reuse). LDS stride 132 floats keeps the
// strided f32-WMMA A-fragment reads bank-conflict free (banks 4m+c / 4m+c+2).
// All loads are index-clamped (no per-iteration EXEC branches); garbage rows
// only affect guarded output rows.
// ---------------------------------------------------------------------------
template<int DIN, int DOUT>
__global__ __launch_bounds__(64)
void gin_mlp_wmma_kernel(const float* __restrict__ x, const float* __restrict__ agg,
                         const float* __restrict__ W1, const float* __restrict__ b1,
                         const float* __restrict__ gamma, const float* __restrict__ beta,
                         const float* __restrict__ mean, const float* __restrict__ var,
                         const float* __restrict__ W2, const float* __restrict__ b2,
                         float* __restrict__ out, int nNodes) {
    static_assert(DIN == 128, "staging sized for DIN=128");
    constexpr int LSTR = DIN + 4;           // 132 floats
    constexpr int NT1  = DIN / 16;          // 8 column tiles for GEMM1
    constexpr int NT2  = DOUT / 16;         // GEMM2 column tiles
    __shared__ float lds[2 * 32 * LSTR];    // 2 waves x 32 rows: 33792 B

    const int lane   = threadIdx.x & 31;
    const int wv     = threadIdx.x >> 5;
    const int m      = lane & 15;           // WMMA row / column within tile
    const int koff   = (lane >> 4) << 1;    // lanes 16-31 carry K+2,K+3
    const int rowoff = (lane >> 4) << 3;    // C-frag: lanes 16-31 hold M=r+8
    float* tile = &lds[wv * 32 * LSTR];     // two 16-row tiles back-to-back
    const int node0 = (blockIdx.x * 2 + wv) * 32;

    // ---- stage h = x + agg (32 x 128, coalesced 512B rows, clamped) ----
    const int c4 = lane * 4;
    for (int r = 0; r < 32; ++r) {
        int node = node0 + r;
        node = node < nNodes ? node : (nNodes - 1);
        const float4 xv = *(const float4*)(x   + (size_t)node * DIN + c4);
        const float4 av = *(const float4*)(agg + (size_t)node * DIN + c4);
        *(float4*)(tile + r * LSTR + c4) =
            make_float4(xv.x + av.x, xv.y + av.y, xv.z + av.z, xv.w + av.w);
    }
    __syncthreads();

    // ---- GEMM1: two (16 x DIN) tiles @ (DIN x DIN), K-step 4 ----
    v8f c1[2][NT1];
#pragma unroll
    for (int t = 0; t < 2; ++t)
#pragma unroll
        for (int i = 0; i < NT1; ++i)
            c1[t][i] = (v8f){0.f,0.f,0.f,0.f,0.f,0.f,0.f,0.f};

    for (int k = 0; k < DIN / 4; ++k) {
        const int krow = k * 4 + koff;
        v2f a0 = *(const v2f*)(tile + m * LSTR + krow);          // tile 0
        v2f a1 = *(const v2f*)(tile + (16 + m) * LSTR + krow);   // tile 1
#pragma unroll
        for (int nt = 0; nt < NT1; ++nt) {
            const int n = nt * 16 + m;
            v2f b;
            b.x = W1[(size_t)krow * DIN + n];
            b.y = W1[(size_t)(krow + 1) * DIN + n];
            c1[0][nt] = __builtin_amdgcn_wmma_f32_16x16x4_f32(
                false, a0, false, b, (short)0, c1[0][nt], false, false);
            c1[1][nt] = __builtin_amdgcn_wmma_f32_16x16x4_f32(
                false, a1, false, b, (short)0, c1[1][nt], false, false);
        }
    }

    // ---- bias + BN(eval) + ReLU, write mid result back into the LDS tiles ----
#pragma unroll
    for (int nt = 0; nt < NT1; ++nt) {
        const int n = nt * 16 + m;
        const float sc = gamma[n] * rsqrtf(var[n] + BN_EPS_C);
        const float sh = (b1[n] - mean[n]) * sc + beta[n];
#pragma unroll
        for (int t = 0; t < 2; ++t)
#pragma unroll
            for (int r = 0; r < 8; ++r) {
                float y = fmaxf(c1[t][nt][r] * sc + sh, 0.f);
                tile[(t * 16 + r + rowoff) * LSTR + n] = y;
            }
    }

    // ---- GEMM2: two (16 x DIN) tiles @ (DIN x DOUT) ----
    v8f c2[2][NT2];
#pragma unroll
    for (int t = 0; t < 2; ++t)
#pragma unroll
        for (int i = 0; i < NT2; ++i)
            c2[t][i] = (v8f){0.f,0.f,0.f,0.f,0.f,0.f,0.f,0.f};

    for (int k = 0; k < DIN / 4; ++k) {
        const int krow = k * 4 + koff;
        v2f a0 = *(const v2f*)(tile + m * LSTR + krow);
        v2f a1 = *(const v2f*)(tile + (16 + m) * LSTR + krow);
#pragma unroll
        for (int nt = 0; nt < NT2; ++nt) {
            const int n = nt * 16 + m;
            v2f b;
            b.x = W2[(size_t)krow * DOUT + n];
            b.y = W2[(size_t)(krow + 1) * DOUT + n];
            c2[0][nt] = __builtin_amdgcn_wmma_f32_16x16x4_f32(
                false, a0, false, b, (short)0, c2[0][nt], false, false);
            c2[1][nt] = __builtin_amdgcn_wmma_f32_16x16x4_f32(
                false, a1, false, b, (short)0, c2[1][nt], false, false);
        }
    }

    // ---- bias + ReLU, guarded store ----
#pragma unroll
    for (int nt = 0; nt < NT2; ++nt) {
        const int n = nt * 16 + m;
        const float bb = b2[n];
#pragma unroll
        for (int t = 0; t < 2; ++t)
#pragma unroll
            for (int r = 0; r < 8; ++r) {
                int node = node0 + t * 16 + r + rowoff;
                if (node < nNodes)
                    out[(size_t)node * DOUT + n] = fmaxf(c2[t][nt][r] + bb, 0.f);
            }
    }
}

// ---------------------------------------------------------------------------
// FC head: logits(N x 32) = emb(N x 64) @ W(64 x 32) + b   (no activation)
// A loads index-clamped -> no per-k EXEC branches.
// ---------------------------------------------------------------------------
__global__ __launch_bounds__(256)
void gin_fc_wmma_kernel(const float* __restrict__ emb, const float* __restrict__ W,
                        const float* __restrict__ b, float* __restrict__ out, int nNodes) {
    const int lane   = threadIdx.x & 31;
    const int wv     = threadIdx.x >> 5;
    const int m      = lane & 15;
    const int koff   = (lane >> 4) << 1;
    const int rowoff = (lane >> 4) << 3;
    const int node0  = (blockIdx.x * 8 + wv) * 16;
    int nodeA = node0 + m;
    nodeA = nodeA < nNodes ? nodeA : (nNodes - 1);
    const float* arow = emb + (size_t)nodeA * 64;

    v8f c[2];
    c[0] = (v8f){0.f,0.f,0.f,0.f,0.f,0.f,0.f,0.f};
    c[1] = c[0];

    for (int k = 0; k < 16; ++k) {          // K = 64
        const int krow = k * 4 + koff;
        v2f a = *(const v2f*)(arow + krow);
#pragma unroll
        for (int nt = 0; nt < 2; ++nt) {
            const int n = nt * 16 + m;
            v2f bf;
            bf.x = W[(size_t)krow * 32 + n];
            bf.y = W[(size_t)(krow + 1) * 32 + n];
            c[nt] = __builtin_amdgcn_wmma_f32_16x16x4_f32(
                false, a, false, bf, (short)0, c[nt], false, false);
        }
    }
#pragma unroll
    for (int nt = 0; nt < 2; ++nt) {
        const int n = nt * 16 + m;
        const float bb = b[n];
#pragma unroll
        for (int r = 0; r < 8; ++r) {
            int node = node0 + r + rowoff;
            if (node < nNodes) out[(size_t)node * 32 + n] = c[nt][r] + bb;
        }
    }
}

// ---------------------------------------------------------------------------
extern "C" void kernel_launch(void* const* d_in, const int* in_sizes, int n_in,
                              void* d_out, int out_size, void* d_ws, size_t ws_size,
                              hipStream_t stream) {
    const int N = N_NODES_C, E = N_EDGES_C;

    const float* x  = (const float*)d_in[0];
    const int*   ei = (const int*)d_in[1];
    // l1 params
    const float *l1W1 = (const float*)d_in[2],  *l1b1 = (const float*)d_in[3];
    const float *l1g  = (const float*)d_in[4],  *l1be = (const float*)d_in[5];
    const float *l1m  = (const float*)d_in[6],  *l1v  = (const float*)d_in[7];
    const float *l1W2 = (const float*)d_in[8],  *l1b2 = (const float*)d_in[9];
    // l2 params
    const float *l2W1 = (const float*)d_in[10], *l2b1 = (const float*)d_in[11];
    const float *l2g  = (const float*)d_in[12], *l2be = (const float*)d_in[13];
    const float *l2m  = (const float*)d_in[14], *l2v  = (const float*)d_in[15];
    const float *l2W2 = (const float*)d_in[16], *l2b2 = (const float*)d_in[17];
    // fc
    const float *fcW  = (const float*)d_in[18], *fcb  = (const float*)d_in[19];

    float* agg = (float*)d_ws;                       // N*128 f32
    float* h1  = agg + (size_t)N * DIM_C;            // N*128 f32
    float* emb    = (float*)d_out;                   // N*64
    float* logits = emb + (size_t)N * 64;            // N*32

    const size_t nAgg4 = (size_t)N * DIM_C / 4;
    const int zeroBlocks = 2048;
    const int scatBlocks = (E + 7) / 8;              // 8 edges (waves) per 256-thr block
    const int mlpBlocks  = (N + 63) / 64;            // 2 waves x 32 nodes per block
    const int fcBlocks   = (N + 127) / 128;

    // ---- layer 1 ----
    gin_zero_kernel<<<zeroBlocks, 256, 0, stream>>>(agg, nAgg4);
    gin_scatter_kernel<<<scatBlocks, 256, 0, stream>>>(x, ei, agg, E);
    gin_mlp_wmma_kernel<128, 128><<<mlpBlocks, 64, 0, stream>>>(
        x, agg, l1W1, l1b1, l1g, l1be, l1m, l1v, l1W2, l1b2, h1, N);

    // ---- layer 2 ----
    gin_zero_kernel<<<zeroBlocks, 256, 0, stream>>>(agg, nAgg4);
    gin_scatter_kernel<<<scatBlocks, 256, 0, stream>>>(h1, ei, agg, E);
    gin_mlp_wmma_kernel<128, 64><<<mlpBlocks, 64, 0, stream>>>(
        h1, agg, l2W1, l2b1, l2g, l2be, l2m, l2v, l2W2, l2b2, emb, N);

    // ---- fc head ----
    gin_fc_wmma_kernel<<<fcBlocks, 256, 0, stream>>>(emb, fcW, fcb, logits, N);
}